// OptNetEq_47244640256492
// MI455X (gfx1250) — compile-verified
//
#include <hip/hip_runtime.h>

#define QPEN  0.1f
#define SIGMA 0.1f
#define N_ITERS 15
#define NFEAT 784
#define NHID  128
#define NCLS  10
#define NEQ   32
#define NINEQ 64
#define BATCH 512
#define LDH   130          // padded row stride for sH (bank-conflict avoidance)

typedef __attribute__((ext_vector_type(2))) float v2f;
typedef __attribute__((ext_vector_type(8))) float v8f;

// ---------------------------------------------------------------------------
// Kernel 1: p = -relu(x @ W1^T + b1)   [512,784]x[784,128] via WMMA f32 16x16x4
// one wave per 16x16 output tile; K = 784 = 196 * 4
// ---------------------------------------------------------------------------
__global__ __launch_bounds__(32) void k1_layer1(const float* __restrict__ x,
                                                const float* __restrict__ W1,
                                                const float* __restrict__ b1,
                                                float* __restrict__ p) {
  const int tile = blockIdx.x;          // 0..255 : 32 batch-tiles x 8 hidden-tiles
  const int tm   = tile >> 3;
  const int tn   = tile & 7;
  const int lane = threadIdx.x;
  const int r    = lane & 15;
  const int half = lane >> 4;
  const int m0   = tm * 16;
  const int n0   = tn * 16;

  v8f c = {};
  const float* xrow = x  + (m0 + r) * NFEAT;   // A row (M = m0+r)
  const float* wrow = W1 + (n0 + r) * NFEAT;   // B col (N = n0+r): B[k][n] = W1[n][k]
  for (int k0 = 0; k0 < NFEAT; k0 += 4) {
    const int k = k0 + 2 * half;               // lanes 0-15: K=k0,k0+1 ; 16-31: K=k0+2,k0+3
    v2f a, bv;
    a.x  = xrow[k];  a.y  = xrow[k + 1];
    bv.x = wrow[k];  bv.y = wrow[k + 1];
    c = __builtin_amdgcn_wmma_f32_16x16x4_f32(false, a, false, bv, (short)0, c,
                                              false, false);
  }
  for (int v = 0; v < 8; ++v) {
    const int row = m0 + v + 8 * half;
    const int col = n0 + r;
    float val = c[v] + b1[col];
    val = val > 0.f ? val : 0.f;
    p[row * NHID + col] = -val;
  }
}

// ---------------------------------------------------------------------------
// Kernel 2: one workgroup (256 threads = 8 waves) per batch element.
// 15-iteration primal-dual IPM entirely in LDS.
//   * H = 0.1 I + G^T diag(w) G          : WMMA f32 16x16x4 (64 tiles)
//   * blocked Cholesky (NB=16)           : trailing update via WMMA (84 tiles)
//   * tri-solves                         : wave-per-column, lane-parallel dots
// ---------------------------------------------------------------------------
__global__ __launch_bounds__(256) void k2_qp(const float* __restrict__ pbuf,
                                             const float* __restrict__ G,
                                             const float* __restrict__ hvec,
                                             const float* __restrict__ A,
                                             const float* __restrict__ beq,
                                             float* __restrict__ zbuf) {
  __shared__ float sH[NHID * LDH];    // 66.5 KB : Schur/ Cholesky factor in place
  __shared__ float sR[NHID * 33];     // 16.9 KB : [A^T | rhs1] -> [HiA | Hir]
  __shared__ float sS[NEQ * 34];      // augmented Schur system [S | rhs]
  __shared__ float sinvd[NHID];       // 1 / L[k][k]
  __shared__ float sp[NHID], sz[NHID], sdz[NHID];
  __shared__ float ss[NINEQ], slam[NINEQ], sw[NINEQ], st1[NINEQ];
  __shared__ float sds[NINEQ], sdlam[NINEQ], sri[NINEQ], red[NINEQ];
  __shared__ float snu[NEQ], sdnu[NEQ], sre[NEQ];
  __shared__ float scal[2];           // [0]=mu  [1]=pivot-inv / alpha

  const int tid  = threadIdx.x;
  const int b    = blockIdx.x;
  const int lane = tid & 31;
  const int wv   = tid >> 5;
  const int r    = lane & 15;
  const int half = lane >> 4;

  if (tid < NHID)  { sp[tid] = pbuf[b * NHID + tid]; sz[tid] = 0.f; }
  if (tid < NINEQ) { ss[tid] = 1.f; slam[tid] = 1.f; }
  if (tid < NEQ)   { snu[tid] = 0.f; }
  __syncthreads();

  for (int it = 0; it < N_ITERS; ++it) {
    // ---- mu = dot(s, lam)/m ------------------------------------------------
    if (tid == 0) {
      float acc = 0.f;
      for (int j = 0; j < NINEQ; ++j) acc += ss[j] * slam[j];
      scal[0] = acc / (float)NINEQ;
    }
    __syncthreads();
    const float mu = scal[0];

    // ---- r_ineq = G z + s - h ; r_eq = A z - b -----------------------------
    if (tid < NINEQ) {
      float acc = ss[tid] - hvec[tid];
      const float* Gr = G + tid * NHID;
      for (int i = 0; i < NHID; ++i) acc += Gr[i] * sz[i];
      sri[tid] = acc;
    } else if (tid < NINEQ + NEQ) {
      const int e = tid - NINEQ;
      float acc = -beq[e];
      const float* Ar = A + e * NHID;
      for (int i = 0; i < NHID; ++i) acc += Ar[i] * sz[i];
      sre[e] = acc;
    }
    __syncthreads();

    // ---- w = lam/s ; t1 = (r_comp - lam*r_ineq)/s --------------------------
    if (tid < NINEQ) {
      const float l = slam[tid], s = ss[tid];
      sw[tid]  = l / s;
      st1[tid] = (l * s - SIGMA * mu - l * sri[tid]) / s;
    }
    __syncthreads();

    // ---- rhs1 = -r_dual + G^T t1 -> sR[:,32] ; A^T -> sR[:,0..31] ----------
    if (tid < NHID) {
      const int i = tid;
      float rd = QPEN * sz[i] + sp[i];
      float gt = 0.f;
      for (int j = 0; j < NINEQ; ++j) {
        const float gji = G[j * NHID + i];
        rd += gji * slam[j];
        gt += gji * st1[j];
      }
      for (int e = 0; e < NEQ; ++e) rd += A[e * NHID + i] * snu[e];
      sR[i * 33 + 32] = gt - rd;
    }
    for (int idx = tid; idx < NHID * NEQ; idx += 256) {
      const int i = idx >> 5, e = idx & 31;
      sR[i * 33 + e] = A[e * NHID + i];
    }
    __syncthreads();

    // ---- H = QPEN*I + G^T diag(w) G  (WMMA f32, 8x8 tiles of 16x16) --------
    for (int t = wv; t < 64; t += 8) {
      const int ti = t >> 3, tj = t & 7;
      v8f c = {};
      for (int k0 = 0; k0 < NINEQ; k0 += 4) {
        const int k = k0 + 2 * half;
        v2f a, bv;
        a.x  = G[k * NHID + ti * 16 + r];            // A[m][k] = G[k][m]
        a.y  = G[(k + 1) * NHID + ti * 16 + r];
        bv.x = G[k * NHID + tj * 16 + r] * sw[k];    // B[k][n] = w[k] G[k][n]
        bv.y = G[(k + 1) * NHID + tj * 16 + r] * sw[k + 1];
        c = __builtin_amdgcn_wmma_f32_16x16x4_f32(false, a, false, bv,
                                                  (short)0, c, false, false);
      }
      for (int v = 0; v < 8; ++v) {
        const int row = ti * 16 + v + 8 * half;
        const int col = tj * 16 + r;
        float val = c[v];
        if (row == col) val += QPEN;
        sH[row * LDH + col] = val;
      }
    }
    __syncthreads();

    // ---- blocked Cholesky, NB=16; trailing Schur update via WMMA -----------
    for (int kb = 0; kb < 8; ++kb) {
      const int k0   = kb * 16;
      const int kend = k0 + 16;
      // panel factor (columns k0..kend-1, all rows below)
      for (int k = k0; k < kend; ++k) {
        const float dsq = sH[k * LDH + k];
        const float inv = 1.0f / sqrtf(dsq);         // redundant per-thread
        for (int i = k + 1 + tid; i < NHID; i += 256) sH[i * LDH + k] *= inv;
        if (tid == 0) sinvd[k] = inv;
        __syncthreads();
        const int nc = kend - 1 - k;                 // panel cols right of k
        if (nc > 0) {
          const int nr = NHID - 1 - k;
          for (int idx = tid; idx < nr * nc; idx += 256) {
            const int i = k + 1 + idx / nc;
            const int j = k + 1 + idx % nc;
            sH[i * LDH + j] -= sH[i * LDH + k] * sH[j * LDH + k];
          }
        }
        __syncthreads();
      }
      // trailing update: H22 -= L21 * L21^T  (lower tiles only), WMMA rank-16
      if (kb < 7) {
        int cnt = 0;
        for (int bi = kb + 1; bi < 8; ++bi) {
          for (int bj = kb + 1; bj <= bi; ++bj) {
            if ((cnt++ & 7) != wv) continue;
            v8f c = {};
            const int rowb = bi * 16, colb = bj * 16;
            for (int kk0 = 0; kk0 < 16; kk0 += 4) {
              const int kk = k0 + kk0 + 2 * half;
              v2f a, bv;
              a.x  = sH[(rowb + r) * LDH + kk];      // L21 tile (row-major)
              a.y  = sH[(rowb + r) * LDH + kk + 1];
              bv.x = sH[(colb + r) * LDH + kk];      // B[k][n] = L21[n][k]
              bv.y = sH[(colb + r) * LDH + kk + 1];
              c = __builtin_amdgcn_wmma_f32_16x16x4_f32(false, a, false, bv,
                                                        (short)0, c, false, false);
            }
            for (int v = 0; v < 8; ++v)
              sH[(rowb + v + 8 * half) * LDH + colb + r] -= c[v];
          }
        }
        __syncthreads();
      }
    }

    // ---- tri-solves L y = R, L^T x = y : one wave per column ---------------
    for (int c = wv; c < 33; c += 8) {
      for (int i = 0; i < NHID; ++i) {              // forward
        float part = 0.f;
        for (int k = lane; k < i; k += 32)
          part += sH[i * LDH + k] * sR[k * 33 + c];
        part += __shfl_xor(part, 16, 32);
        part += __shfl_xor(part, 8, 32);
        part += __shfl_xor(part, 4, 32);
        part += __shfl_xor(part, 2, 32);
        part += __shfl_xor(part, 1, 32);
        if (lane == 0) sR[i * 33 + c] = (sR[i * 33 + c] - part) * sinvd[i];
        __builtin_amdgcn_wave_barrier();
      }
      for (int i = NHID - 1; i >= 0; --i) {         // backward (L^T)
        float part = 0.f;
        for (int k = i + 1 + lane; k < NHID; k += 32)
          part += sH[k * LDH + i] * sR[k * 33 + c];
        part += __shfl_xor(part, 16, 32);
        part += __shfl_xor(part, 8, 32);
        part += __shfl_xor(part, 4, 32);
        part += __shfl_xor(part, 2, 32);
        part += __shfl_xor(part, 1, 32);
        if (lane == 0) sR[i * 33 + c] = (sR[i * 33 + c] - part) * sinvd[i];
        __builtin_amdgcn_wave_barrier();
      }
    }
    __syncthreads();

    // ---- S = A HiA ; rhs = A Hir + r_eq ------------------------------------
    for (int idx = tid; idx < NEQ * NEQ; idx += 256) {
      const int e = idx >> 5, f = idx & 31;
      float acc = 0.f;
      const float* Ar = A + e * NHID;
      for (int i = 0; i < NHID; ++i) acc += Ar[i] * sR[i * 33 + f];
      sS[e * 34 + f] = acc;
    }
    if (tid < NEQ) {
      float acc = sre[tid];
      const float* Ar = A + tid * NHID;
      for (int i = 0; i < NHID; ++i) acc += Ar[i] * sR[i * 33 + 32];
      sS[tid * 34 + 32] = acc;
    }
    __syncthreads();

    // ---- solve S dnu = rhs  (Gaussian elimination, SPD) --------------------
    for (int k = 0; k < NEQ; ++k) {
      if (tid == 0) scal[1] = 1.f / sS[k * 34 + k];
      __syncthreads();
      if (tid > k && tid < NEQ) {
        const float f = sS[tid * 34 + k] * scal[1];
        for (int c = k; c <= NEQ; ++c) sS[tid * 34 + c] -= f * sS[k * 34 + c];
      }
      __syncthreads();
    }
    if (tid == 0) {
      for (int i = NEQ - 1; i >= 0; --i) {
        float acc = sS[i * 34 + 32];
        for (int k = i + 1; k < NEQ; ++k) acc -= sS[i * 34 + k] * sdnu[k];
        sdnu[i] = acc / sS[i * 34 + i];
      }
    }
    __syncthreads();

    // ---- dz = Hir - HiA dnu ------------------------------------------------
    if (tid < NHID) {
      float acc = sR[tid * 33 + 32];
      for (int f = 0; f < NEQ; ++f) acc -= sR[tid * 33 + f] * sdnu[f];
      sdz[tid] = acc;
    }
    __syncthreads();

    // ---- ds, dlam, fraction-to-boundary ------------------------------------
    if (tid < NINEQ) {
      float acc = -sri[tid];
      const float* Gr = G + tid * NHID;
      for (int i = 0; i < NHID; ++i) acc -= Gr[i] * sdz[i];
      const float dsj = acc;
      sds[tid] = dsj;
      const float l = slam[tid], s = ss[tid];
      const float rcomp = l * s - SIGMA * mu;
      const float dlj = (-rcomp - l * dsj) / s;
      sdlam[tid] = dlj;
      const float a1 = dsj < 0.f ? -s / dsj : 3.4e38f;
      const float a2 = dlj < 0.f ? -l / dlj : 3.4e38f;
      red[tid] = fminf(a1, a2);
    }
    __syncthreads();
    if (tid == 0) {
      float m = 3.4e38f;
      for (int j = 0; j < NINEQ; ++j) m = fminf(m, red[j]);
      scal[1] = fminf(1.f, 0.99f * m);
    }
    __syncthreads();
    const float alpha = scal[1];
    if (tid < NHID)  sz[tid] += alpha * sdz[tid];
    if (tid < NINEQ) { ss[tid] += alpha * sds[tid]; slam[tid] += alpha * sdlam[tid]; }
    if (tid < NEQ)   snu[tid] += alpha * sdnu[tid];
    __syncthreads();
  }

  if (tid < NHID) zbuf[b * NHID + tid] = sz[tid];
}

// ---------------------------------------------------------------------------
// Kernel 3: logits = z @ W2^T + b2 ; out = log_softmax(logits)
// ---------------------------------------------------------------------------
__global__ __launch_bounds__(128) void k3_head(const float* __restrict__ zbuf,
                                               const float* __restrict__ W2,
                                               const float* __restrict__ b2,
                                               float* __restrict__ out) {
  __shared__ float zz[NHID];
  __shared__ float lg[NCLS];
  __shared__ float lse;
  const int b = blockIdx.x, tid = threadIdx.x;
  if (tid < NHID) zz[tid] = zbuf[b * NHID + tid];
  __syncthreads();
  if (tid < NCLS) {
    float acc = b2[tid];
    const float* w = W2 + tid * NHID;
    for (int i = 0; i < NHID; ++i) acc += w[i] * zz[i];
    lg[tid] = acc;
  }
  __syncthreads();
  if (tid == 0) {
    float mx = lg[0];
    for (int c = 1; c < NCLS; ++c) mx = fmaxf(mx, lg[c]);
    float se = 0.f;
    for (int c = 0; c < NCLS; ++c) se += expf(lg[c] - mx);
    lse = mx + logf(se);
  }
  __syncthreads();
  if (tid < NCLS) out[b * NCLS + tid] = lg[tid] - lse;
}

// ---------------------------------------------------------------------------
extern "C" void kernel_launch(void* const* d_in, const int* in_sizes, int n_in,
                              void* d_out, int out_size, void* d_ws, size_t ws_size,
                              hipStream_t stream) {
  const float* x   = (const float*)d_in[0];
  const float* W1  = (const float*)d_in[1];
  const float* b1  = (const float*)d_in[2];
  const float* W2  = (const float*)d_in[3];
  const float* b2  = (const float*)d_in[4];
  const float* G   = (const float*)d_in[5];
  const float* h   = (const float*)d_in[6];
  const float* A   = (const float*)d_in[7];
  const float* beq = (const float*)d_in[8];

  float* pbuf = (float*)d_ws;                 // [512,128]
  float* zbuf = pbuf + BATCH * NHID;          // [512,128]
  float* out  = (float*)d_out;                // [512,10] fp32

  k1_layer1<<<dim3((BATCH / 16) * (NHID / 16)), dim3(32), 0, stream>>>(x, W1, b1, pbuf);
  k2_qp<<<dim3(BATCH), dim3(256), 0, stream>>>(pbuf, G, h, A, beq, zbuf);
  k3_head<<<dim3(BATCH), dim3(128), 0, stream>>>(zbuf, W2, b2, out);
}